// CorrSim_635655160223
// MI455X (gfx1250) — compile-verified
//
#include <hip/hip_runtime.h>
#include <hip/hip_bf16.h>

// Problem constants (match reference setup_inputs)
#define B_DIM 256
#define C_DIM 256
#define HW    256
#define N_SUP 64
#define S_SUP 5

typedef __attribute__((ext_vector_type(2))) float v2f;
typedef __attribute__((ext_vector_type(8))) float v8f;

// ---------------------------------------------------------------------------
// Kernel 1: support statistics from x2.
// Logical channel j = s*C + c  maps to  x2[n, j%5, j/5, k].
// Per (n,k):  m[s] = mean_c X[n, s*C+c, k]
//             s_sum[c] = sum_s (X - m[s])
//             ssum_c = sum_c s_sum ; w2 = sum_c s_sum^2
// Thread owns one k column -> all loads coalesced across lanes.
// ---------------------------------------------------------------------------
__global__ __launch_bounds__(64) void support_kernel(
    const float* __restrict__ x2, float* __restrict__ ssum_c_out,
    float* __restrict__ w2_out) {
  const int n = blockIdx.x;
  const int k = blockIdx.y * 64 + threadIdx.x;                 // 0..255
  const float* base = x2 + (size_t)n * (S_SUP * C_DIM * HW) + k;

  // ---- pass 1: per-s means over c (each thread for its own k) ----
  float m[S_SUP];
#pragma unroll
  for (int s = 0; s < S_SUP; ++s) {
    int j = s * C_DIM;
    int s0 = j % S_SUP;
    int c0 = j / S_SUP;
    float acc = 0.0f;
#pragma unroll 4
    for (int c = 0; c < C_DIM; ++c) {
      acc += base[(s0 * C_DIM + c0) * HW];
      if (++s0 == S_SUP) { s0 = 0; ++c0; }
    }
    m[s] = acc * (1.0f / (float)C_DIM);
  }

  // ---- pass 2: centered sums (L2-resident re-read) ----
  // Track (s0,c0) for j = c incrementally; advancing s adds 256 = 51*5 + 1.
  float sacc = 0.0f, wacc = 0.0f;
  int s0b = 0, c0b = 0;  // decomposition of j = c
  for (int c = 0; c < C_DIM; ++c) {
    int s0 = s0b, c0 = c0b;
    float ss = 0.0f;
#pragma unroll
    for (int s = 0; s < S_SUP; ++s) {
      ss += base[(s0 * C_DIM + c0) * HW] - m[s];
      if (++s0 == S_SUP) { s0 = 0; c0 += 52; } else { c0 += 51; }
    }
    sacc += ss;
    wacc += ss * ss;
    if (++s0b == S_SUP) { s0b = 0; ++c0b; }
  }
  ssum_c_out[n * HW + k] = sacc;
  w2_out[n * HW + k] = wacc;
}

// ---------------------------------------------------------------------------
// Kernel 2: query statistics from x1.
// Phase 1: rnorm[c] = 1/||x1[b,c,:]|| via wave32 shuffle reduction -> LDS.
// Phase 2: qsum[b,k] = sum_c x1[b,c,k]*rnorm[c] using
//          v_wmma_f32_16x16x4_f32 with A = ones (exact ones-vector GEMV;
//          invariant to the K-slot permutation inside B's VGPR layout).
// ---------------------------------------------------------------------------
__global__ __launch_bounds__(256) void query_kernel(
    const float* __restrict__ x1, float* __restrict__ qsum) {
  const int b = blockIdx.x;
  const float* xb = x1 + (size_t)b * (C_DIM * HW);
  __shared__ float rnorm[C_DIM];

  const int tid  = threadIdx.x;
  const int wave = tid >> 5;
  const int lane = tid & 31;

  // ---- phase 1: row inverse norms (wave w handles rows 32w..32w+31) ----
  for (int r = 0; r < 32; ++r) {
    const int c = wave * 32 + r;
    const float* row = xb + c * HW;
    float ss = 0.0f;
#pragma unroll
    for (int i = 0; i < 8; ++i) {
      float v = row[lane + 32 * i];
      ss += v * v;
    }
#pragma unroll
    for (int off = 16; off > 0; off >>= 1) ss += __shfl_xor(ss, off, 32);
    if (lane == 0) rnorm[c] = 1.0f / sqrtf(ss);
  }
  __syncthreads();

  // ---- phase 2: WMMA ones-reduction over channels ----
  // B[kk, n] = x1[b, cbase+kk, kbase+n] * rnorm[cbase+kk]
  // lane = n + 16*hi ; this lane's two B VGPRs carry channels
  // cbase+2*hi and cbase+2*hi+1 (any kk permutation sums identically).
  v2f ones;
  ones.x = 1.0f;
  ones.y = 1.0f;
  const int ncol = lane & 15;
  const int hi   = lane >> 4;

  for (int g = 0; g < 2; ++g) {
    const int kbase = (wave * 2 + g) * 16;
    v8f acc = {};  // C starts at zero
    for (int cbase = 0; cbase < C_DIM; cbase += 4) {
      const int ca = cbase + 2 * hi;
      v2f bv;
      bv.x = xb[ca * HW + kbase + ncol] * rnorm[ca];
      bv.y = xb[(ca + 1) * HW + kbase + ncol] * rnorm[ca + 1];
      acc = __builtin_amdgcn_wmma_f32_16x16x4_f32(
          /*neg_a=*/false, ones, /*neg_b=*/false, bv,
          /*c_mod=*/(short)0, acc, /*reuse_a=*/false, /*reuse_b=*/false);
    }
    // All 16 rows of D are identical; lanes 0..15 hold columns 0..15.
    if (hi == 0) qsum[b * HW + kbase + ncol] = acc[0];
  }
}

// ---------------------------------------------------------------------------
// Kernel 3: elementwise combine -> out[b, n*hw + k]
// sim = qsum*ssum_c / sqrt(max(C*qsum^2 * w2, EPS^2)), EPS = 1e-8
// ---------------------------------------------------------------------------
__global__ __launch_bounds__(256) void combine_kernel(
    const float* __restrict__ qsum, const float* __restrict__ ssum_c,
    const float* __restrict__ w2, float* __restrict__ out) {
  const int b = blockIdx.x;
  const int k = threadIdx.x;
  const float qv = qsum[b * HW + k];
  const float w1 = (float)C_DIM * qv * qv;
  float* ob = out + (size_t)b * (N_SUP * HW) + k;
#pragma unroll 4
  for (int n = 0; n < N_SUP; ++n) {
    const float sc = ssum_c[n * HW + k];
    const float wv = w2[n * HW + k];
    const float denom = sqrtf(fmaxf(w1 * wv, 1e-16f));
    ob[n * HW] = qv * sc / denom;
  }
}

extern "C" void kernel_launch(void* const* d_in, const int* in_sizes, int n_in,
                              void* d_out, int out_size, void* d_ws,
                              size_t ws_size, hipStream_t stream) {
  const float* x1 = (const float*)d_in[0];  // [256,256,16,16]
  const float* x2 = (const float*)d_in[1];  // [64,5,256,16,16]
  float* out = (float*)d_out;               // [256, 64*256]

  float* ws     = (float*)d_ws;
  float* ssum_c = ws;                       // N*HW   = 16384 floats
  float* w2     = ws + N_SUP * HW;          // N*HW   = 16384 floats
  float* qsum   = ws + 2 * N_SUP * HW;      // B*HW   = 65536 floats

  support_kernel<<<dim3(N_SUP, 4), 64, 0, stream>>>(x2, ssum_c, w2);
  query_kernel<<<B_DIM, 256, 0, stream>>>(x1, qsum);
  combine_kernel<<<B_DIM, 256, 0, stream>>>(qsum, ssum_c, w2, out);
}